// _PSRoIPooling_49658411876954
// MI455X (gfx1250) — compile-verified
//
#include <hip/hip_runtime.h>

// ---------------- problem constants (from reference) ----------------
#define HH    96
#define WW    160
#define LDSW  161          // odd stride -> conflict-free column access on 64 banks
#define CTOT  1029         // DOUT*G*G = 21*49
#define NROIS 1024
#define NBAT  2

typedef __attribute__((ext_vector_type(2))) float    v2f;
typedef __attribute__((ext_vector_type(8))) float    v8f;
typedef __attribute__((ext_vector_type(4))) unsigned v4u;
typedef __attribute__((ext_vector_type(8))) unsigned v8u;

static __device__ __forceinline__ v8f wmma_f32_k4(v2f a, v2f b, v8f c) {
  // D(16x16,f32) = A(16x4,f32) x B(4x16,f32) + C   -- full fp32 matrix op
  return __builtin_amdgcn_wmma_f32_16x16x4_f32(false, a, false, b, (short)0, c,
                                               false, false);
}

static __device__ __forceinline__ float bcast_lane(float v, int srcLane) {
  return __builtin_bit_cast(
      float, __builtin_amdgcn_ds_bpermute(srcLane << 2,
                                          __builtin_bit_cast(int, v)));
}

__global__ __launch_bounds__(256) void psroi_fused_kernel(
    const float* __restrict__ feat, const float* __restrict__ rois,
    float* __restrict__ out) {
  // SAT plane in LDS, stride 161 (only 96x160 used; zero border is virtual)
  __shared__ float sat[HH * LDSW];

  const int plane = blockIdx.x;          // = n*CTOT + c
  const int n     = plane / CTOT;
  const int c     = plane - n * CTOT;
  const int t     = threadIdx.x;
  const int lane  = t & 31;
  const int wave  = __builtin_amdgcn_readfirstlane(t >> 5);
  const int hl    = lane >> 4;           // 0 = lanes 0-15, 1 = lanes 16-31
  const int l15   = lane & 15;

  // ---------------- Phase 1: TDM fill (HBM -> LDS), one DMA per block ------
  if (wave == 0) {
    unsigned long long ga =
        (unsigned long long)(const void*)(feat + (size_t)plane * (HH * WW));
    unsigned la = (unsigned)(unsigned long long)(const void*)&sat[0];
    v4u g0; v8u g1; v4u g2; v4u g3;
    // D# group0: count=1, lds_addr, 57b global addr, type=2 ("image")
    g0[0] = 1u;
    g0[1] = la;
    g0[2] = (unsigned)ga;
    g0[3] = (unsigned)(ga >> 32) | (2u << 30);
    // D# group1: data_size=4B (2), iterate_enable, dims/strides, 160x1 tile
    g1[0] = (2u << 16) | (1u << 19);
    g1[1] = (unsigned)WW << 16;        // tensor_dim0[15:0]
    g1[2] = (unsigned)HH << 16;        // tensor_dim0[31:16]=0 | tensor_dim1[15:0]
    g1[3] = (unsigned)WW << 16;        // tensor_dim1[31:16]=0 | tile_dim0=160
    g1[4] = 1u;                        // tile_dim1=1, tile_dim2=0
    g1[5] = (unsigned)WW;              // tensor_dim0_stride = 160 elements
    g1[6] = 0u;
    g1[7] = 0u;
    // D# group2 (iterate mode): lds += 161 elems, global += 160 elems, 96 iters
    g2[0] = 0u;
    g2[1] = (unsigned)LDSW;            // lds_addr_increment
    g2[2] = (unsigned)WW;              // global_addr_increment[31:0]
    g2[3] = (unsigned)(HH - 1) << 16;  // iterate_count = 95 -> 96 rows
    g3[0] = g3[1] = g3[2] = g3[3] = 0u;
    asm volatile("tensor_load_to_lds %0, %1, %2, %3"
                 :: "s"(g0), "s"(g1), "s"(g2), "s"(g3) : "memory");
    __builtin_amdgcn_s_wait_tensorcnt(0);
  }
  __syncthreads();

  // Triangular operand chunks, built in registers from lane id.
  // Row pass B_k = U[4k:4k+4, :] (upper-tri ones); identical register image
  // serves as col pass A_k = L[:, 4k:4k+4] (lower-tri ones) by layout symmetry.
  v2f tri[4];
#pragma unroll
  for (int k = 0; k < 4; ++k) {
    const int r0 = 4 * k + 2 * hl;
    tri[k].x = (r0     <= l15) ? 1.0f : 0.0f;
    tri[k].y = (r0 + 1 <= l15) ? 1.0f : 0.0f;
  }

  // ---------------- Phase 2: row-wise prefix (along W) via WMMA ------------
  // 6 row strips of 16 rows; wave w owns strip w (waves 6,7 idle here).
  if (wave < 6) {
    const int rbase = wave * 16;
    const int arow  = rbase + l15;
    float c8[8];
#pragma unroll
    for (int j = 0; j < 8; ++j) c8[j] = 0.0f;
    for (int tc = 0; tc < 10; ++tc) {
      v8f acc;
#pragma unroll
      for (int j = 0; j < 8; ++j) acc[j] = c8[j];    // carry-in as C matrix
#pragma unroll
      for (int k = 0; k < 4; ++k) {                  // Y = X @ U (K=16 in 4 chunks)
        const int col = tc * 16 + 4 * k + 2 * hl;
        v2f a;
        a.x = sat[arow * LDSW + col];
        a.y = sat[arow * LDSW + col + 1];
        acc = wmma_f32_k4(a, tri[k], acc);
      }
#pragma unroll
      for (int j = 0; j < 8; ++j) {
        sat[(rbase + j + 8 * hl) * LDSW + tc * 16 + l15] = acc[j];
        // per-row running total = column 15, half-wave broadcast
        c8[j] = bcast_lane(acc[j], (lane & 16) | 15);
      }
    }
  }
  __syncthreads();

  // ---------------- Phase 3: column-wise prefix (along H) via WMMA ---------
  // 10 column strips; wave w owns strips w and w+8.
  for (int cs = wave; cs < 10; cs += 8) {
    const int col = cs * 16 + l15;
    float carry = 0.0f;
    for (int tr = 0; tr < 6; ++tr) {
      v8f acc;
#pragma unroll
      for (int j = 0; j < 8; ++j) acc[j] = carry;    // per-column carry-in
#pragma unroll
      for (int k = 0; k < 4; ++k) {                  // Y = L @ X
        const int row = tr * 16 + 4 * k + 2 * hl;
        v2f b;
        b.x = sat[row * LDSW + col];
        b.y = sat[(row + 1) * LDSW + col];
        acc = wmma_f32_k4(tri[k], b, acc);
      }
#pragma unroll
      for (int j = 0; j < 8; ++j)
        sat[(tr * 16 + j + 8 * hl) * LDSW + col] = acc[j];
      // carry = last row (M=15) of tile, broadcast per column to all lanes
      carry = bcast_lane(acc[7], 16 + l15);
    }
  }
  __syncthreads();

  // ---------------- Phase 4: PSRoI pooling from in-LDS SAT -----------------
  const int ph = (c / 7) % 7;
  const int pw = c % 7;
  for (int r = t; r < NROIS; r += 256) {
    const float* roi = rois + r * 5;
    if ((int)roi[0] != n) continue;                  // handled by other batch's block
    const float x1 = rintf(roi[1]) * 0.0625f;
    const float y1 = rintf(roi[2]) * 0.0625f;
    const float x2 = rintf(roi[3] + 1.0f) * 0.0625f;
    const float y2 = rintf(roi[4] + 1.0f) * 0.0625f;
    const float bin_w = fmaxf(x2 - x1, 0.1f) / 7.0f;
    const float bin_h = fmaxf(y2 - y1, 0.1f) / 7.0f;
    int hs = (int)floorf((float)ph * bin_h + y1);
    int he = (int)ceilf((float)(ph + 1) * bin_h + y1);
    int ws = (int)floorf((float)pw * bin_w + x1);
    int we = (int)ceilf((float)(pw + 1) * bin_w + x1);
    hs = min(max(hs, 0), HH); he = min(max(he, 0), HH);
    ws = min(max(ws, 0), WW); we = min(max(we, 0), WW);
    const int cnt = (he - hs) * (we - ws);
    float v = 0.0f;
    if (cnt > 0) {
      // S[h][w] with virtual zero row/col 0 -> sat[(h-1)][(w-1)]
      auto S = [&](int h, int w) -> float {
        return (h > 0 && w > 0) ? sat[(h - 1) * LDSW + (w - 1)] : 0.0f;
      };
      v = (S(he, we) - S(hs, we) - S(he, ws) + S(hs, ws)) / (float)cnt;
    }
    out[(size_t)r * CTOT + c] = v;
  }
}

extern "C" void kernel_launch(void* const* d_in, const int* in_sizes, int n_in,
                              void* d_out, int out_size, void* d_ws,
                              size_t ws_size, hipStream_t stream) {
  (void)in_sizes; (void)n_in; (void)out_size; (void)d_ws; (void)ws_size;
  const float* feat = (const float*)d_in[0];
  const float* rois = (const float*)d_in[1];
  float* out = (float*)d_out;
  psroi_fused_kernel<<<dim3(NBAT * CTOT), dim3(256), 0, stream>>>(feat, rois, out);
}